// FilterByOctaves_31696858644856
// MI455X (gfx1250) — compile-verified
//
#include <hip/hip_runtime.h>

typedef __attribute__((ext_vector_type(2))) float v2f;
typedef __attribute__((ext_vector_type(8))) float v8f;

// ---------------- problem constants ----------------
constexpr int NBANDS = 6;
constexpr int NSEC   = 3;
constexpr int BATCH  = 64;
constexpr int TLEN   = 96000;

constexpr int BLK       = 16;              // samples per WMMA block
constexpr int BPC       = 60;              // blocks per chunk
constexpr int CHUNK_LEN = BLK * BPC;       // 960
constexpr int CHUNKS    = TLEN / CHUNK_LEN;// 100
constexpr int BTILES    = BATCH / 16;      // 4

// ---------------- workspace layout (floats) ----------------
// per-band parameter block
constexpr int PB       = 1024;
constexpr int TOEP_OFF = 0;    // [16][16] lower-tri Toeplitz of impulse response
constexpr int OBS_OFF  = 256;  // [16][8]  Obs (cols 6,7 = 0)
constexpr int KM_OFF   = 384;  // [16][16] K   (rows 6..15 = 0)
constexpr int PHIA_OFF = 640;  // [16][8]  Phi = A^16 (rows>=6, cols>=6 = 0)
constexpr int PHIC_OFF = 768;  // [6][6]   PhiChunk = A^(960)
constexpr int PARAMS_FLOATS = NBANDS * PB;                 // 6144
constexpr int VSTRIDE  = 96;                               // 6x16 state tile
constexpr int VTOTAL   = NBANDS * BTILES * CHUNKS * VSTRIDE; // 230400
constexpr int V_OFF    = PARAMS_FLOATS;
constexpr int S0_OFF   = V_OFF + VTOTAL;

// ---------------- WMMA helper ----------------
__device__ inline v8f wmma4(v2f a, v2f b, v8f c) {
  // D(16x16,f32) = A(16x4,f32) * B(4x16,f32) + C
  return __builtin_amdgcn_wmma_f32_16x16x4_f32(
      /*neg_a=*/false, a, /*neg_b=*/false, b,
      /*c_mod=*/(short)0, c, /*reuse_a=*/false, /*reuse_b=*/false);
}

// ---------------- setup: derive block state-space matrices from sos ----------------
__global__ void sos_setup_kernel(const float* __restrict__ sos, float* __restrict__ ws) {
  const int f = threadIdx.x;
  if (f >= NBANDS) return;
  float b0[NSEC], b1[NSEC], b2[NSEC], a1[NSEC], a2[NSEC];
  for (int j = 0; j < NSEC; ++j) {
    const float* c = sos + (f * NSEC + j) * 6;
    b0[j] = c[0]; b1[j] = c[1]; b2[j] = c[2]; a1[j] = c[4]; a2[j] = c[5];
  }
  float* P = ws + f * PB;
  for (int i = 0; i < PB; ++i) P[i] = 0.0f;

  auto step = [&](float z[NSEC][2], float xin) -> float {
    float cur = xin;
    for (int j = 0; j < NSEC; ++j) {
      float y   = b0[j] * cur + z[j][0];
      float nz1 = b1[j] * cur - a1[j] * y + z[j][1];
      float nz2 = b2[j] * cur - a2[j] * y;
      z[j][0] = nz1; z[j][1] = nz2;
      cur = y;
    }
    return cur;
  };

  // Impulse run: h[m] and state trajectory (A^m b)
  float h[BLK];
  float straj[BLK + 1][6];
  {
    float z[NSEC][2] = {};
    for (int m = 0; m < BLK; ++m) {
      h[m] = step(z, (m == 0) ? 1.0f : 0.0f);
      for (int i = 0; i < 6; ++i) straj[m + 1][i] = z[i >> 1][i & 1];
    }
  }
  // Toeplitz [time j][time k]
  for (int j = 0; j < BLK; ++j)
    for (int k = 0; k <= j; ++k)
      P[TOEP_OFF + j * 16 + k] = h[j - k];
  // K [state i][time k]: state after block for delta at k = A^(15-k) b = straj[16-k]
  for (int i = 0; i < 6; ++i)
    for (int k = 0; k < BLK; ++k)
      P[KM_OFF + i * 16 + k] = straj[BLK - k][i];
  // Basis-state runs: Obs [time][state], Phi [state'][state]
  float phi6[6][6];
  for (int i = 0; i < 6; ++i) {
    float z[NSEC][2] = {};
    z[i >> 1][i & 1] = 1.0f;
    for (int m = 0; m < BLK; ++m)
      P[OBS_OFF + m * 8 + i] = step(z, 0.0f);
    for (int ii = 0; ii < 6; ++ii) {
      phi6[ii][i] = z[ii >> 1][ii & 1];
      P[PHIA_OFF + ii * 8 + i] = phi6[ii][i];
    }
  }
  // PhiChunk = phi6^BPC (6x6 repeated multiply)
  float acc[6][6], tmp[6][6];
  for (int i = 0; i < 6; ++i) for (int j = 0; j < 6; ++j) acc[i][j] = phi6[i][j];
  for (int p = 1; p < BPC; ++p) {
    for (int i = 0; i < 6; ++i)
      for (int j = 0; j < 6; ++j) {
        float s = 0.0f;
        for (int k = 0; k < 6; ++k) s += phi6[i][k] * acc[k][j];
        tmp[i][j] = s;
      }
    for (int i = 0; i < 6; ++i) for (int j = 0; j < 6; ++j) acc[i][j] = tmp[i][j];
  }
  for (int i = 0; i < 6; ++i) for (int j = 0; j < 6; ++j)
    P[PHIC_OFF + i * 6 + j] = acc[i][j];
}

// ---------------- phase 1: per-chunk zero-state final state (WMMA) ----------------
__global__ __launch_bounds__(32) void phase1_kernel(const float* __restrict__ x,
                                                    float* __restrict__ ws) {
  const int chunk = blockIdx.x, bt = blockIdx.y, f = blockIdx.z;
  const int lane = threadIdx.x;
  const int n = lane & 15, h = lane >> 4;
  const float* P = ws + f * PB;

  v2f KmA[4], PhiA[2];
  for (int c = 0; c < 4; ++c) {
    const int k0 = 4 * c + 2 * h;
    KmA[c].x = P[KM_OFF + n * 16 + k0];
    KmA[c].y = P[KM_OFF + n * 16 + k0 + 1];
  }
  for (int q = 0; q < 2; ++q) {
    const int k0 = 4 * q + 2 * h;
    PhiA[q].x = P[PHIA_OFF + n * 8 + k0];
    PhiA[q].y = P[PHIA_OFF + n * 8 + k0 + 1];
  }

  const float* xrow = x + (size_t)(bt * 16 + n) * TLEN + chunk * CHUNK_LEN;
  float s0 = 0, s1 = 0, s2 = 0, s3 = 0, s4 = 0, s5 = 0;

  for (int blk = 0; blk < BPC; ++blk) {
    const float* xp = xrow + blk * BLK + 2 * h;
    v2f xb[4];
    for (int c = 0; c < 4; ++c) { xb[c].x = xp[4 * c]; xb[c].y = xp[4 * c + 1]; }

    const float s2s = __shfl(s2, n, 32);
    const float s3s = __shfl(s3, n, 32);
    v2f sb0, sb1;
    sb0.x = h ? s2s : s0;  sb0.y = h ? s3s : s1;
    sb1.x = h ? 0.f : s4;  sb1.y = h ? 0.f : s5;

    v8f sn = {};
    sn = wmma4(PhiA[0], sb0, sn);
    sn = wmma4(PhiA[1], sb1, sn);
    for (int c = 0; c < 4; ++c) sn = wmma4(KmA[c], xb[c], sn);
    s0 = sn[0]; s1 = sn[1]; s2 = sn[2]; s3 = sn[3]; s4 = sn[4]; s5 = sn[5];
  }

  float* V = ws + V_OFF + (size_t)((f * BTILES + bt) * CHUNKS + chunk) * VSTRIDE;
  if (h == 0) {
    V[0 * 16 + n] = s0; V[1 * 16 + n] = s1; V[2 * 16 + n] = s2;
    V[3 * 16 + n] = s3; V[4 * 16 + n] = s4; V[5 * 16 + n] = s5;
  }
}

// ---------------- phase 2: sequential 6-dim scan over chunks ----------------
__global__ __launch_bounds__(32) void phase2_kernel(float* __restrict__ ws) {
  const int bt = blockIdx.x, f = blockIdx.y;
  const int lane = threadIdx.x;
  if (lane >= 16) return;
  const int n = lane;
  float pc[36];
  {
    const float* Phic = ws + f * PB + PHIC_OFF;
    for (int i = 0; i < 36; ++i) pc[i] = Phic[i];
  }
  const float* V  = ws + V_OFF  + (size_t)((f * BTILES + bt) * CHUNKS) * VSTRIDE;
  float*       S0 = ws + S0_OFF + (size_t)((f * BTILES + bt) * CHUNKS) * VSTRIDE;
  float s[6] = {};
  for (int c = 0; c < CHUNKS; ++c) {
    for (int i = 0; i < 6; ++i) S0[c * VSTRIDE + i * 16 + n] = s[i];
    float ns[6];
    for (int i = 0; i < 6; ++i) {
      float acc = V[c * VSTRIDE + i * 16 + n];
      for (int j = 0; j < 6; ++j) acc += pc[i * 6 + j] * s[j];
      ns[i] = acc;
    }
    for (int i = 0; i < 6; ++i) s[i] = ns[i];
  }
}

// ---------------- phase 3: full block filter with correct init state (WMMA) ----------------
__global__ __launch_bounds__(32) void phase3_kernel(const float* __restrict__ x,
                                                    const float* __restrict__ ws,
                                                    float* __restrict__ out) {
  __shared__ alignas(16) float tile[16 * 20];  // [batch][time], padded rows
  const int chunk = blockIdx.x, bt = blockIdx.y, f = blockIdx.z;
  const int lane = threadIdx.x;
  const int n = lane & 15, h = lane >> 4;
  const float* P = ws + f * PB;

  v2f Ta[4], KmA[4], ObsA[2], PhiA[2];
  for (int c = 0; c < 4; ++c) {
    const int k0 = 4 * c + 2 * h;
    Ta[c].x  = P[TOEP_OFF + n * 16 + k0]; Ta[c].y  = P[TOEP_OFF + n * 16 + k0 + 1];
    KmA[c].x = P[KM_OFF   + n * 16 + k0]; KmA[c].y = P[KM_OFF   + n * 16 + k0 + 1];
  }
  for (int q = 0; q < 2; ++q) {
    const int k0 = 4 * q + 2 * h;
    ObsA[q].x = P[OBS_OFF  + n * 8 + k0]; ObsA[q].y = P[OBS_OFF  + n * 8 + k0 + 1];
    PhiA[q].x = P[PHIA_OFF + n * 8 + k0]; PhiA[q].y = P[PHIA_OFF + n * 8 + k0 + 1];
  }

  // initial state for this chunk (uniform loads + per-lane select: no divergence)
  const float* S0 = ws + S0_OFF + (size_t)((f * BTILES + bt) * CHUNKS + chunk) * VSTRIDE;
  float t0v = S0[0 * 16 + n], t1v = S0[1 * 16 + n], t2v = S0[2 * 16 + n];
  float t3v = S0[3 * 16 + n], t4v = S0[4 * 16 + n], t5v = S0[5 * 16 + n];
  float s0 = h ? 0.f : t0v, s1 = h ? 0.f : t1v, s2 = h ? 0.f : t2v;
  float s3 = h ? 0.f : t3v, s4 = h ? 0.f : t4v, s5 = h ? 0.f : t5v;

  const float* xrow = x + (size_t)(bt * 16 + n) * TLEN + chunk * CHUNK_LEN;

  for (int blk = 0; blk < BPC; ++blk) {
    const float* xp = xrow + blk * BLK + 2 * h;
    v2f xb[4];
    for (int c = 0; c < 4; ++c) { xb[c].x = xp[4 * c]; xb[c].y = xp[4 * c + 1]; }

    const float s2s = __shfl(s2, n, 32);
    const float s3s = __shfl(s3, n, 32);
    v2f sb0, sb1;
    sb0.x = h ? s2s : s0;  sb0.y = h ? s3s : s1;
    sb1.x = h ? 0.f : s4;  sb1.y = h ? 0.f : s5;

    // Y = Obs*S + Toeplitz*X   (6 chained WMMAs)
    v8f y = {};
    y = wmma4(ObsA[0], sb0, y);
    y = wmma4(ObsA[1], sb1, y);
    for (int c = 0; c < 4; ++c) y = wmma4(Ta[c], xb[c], y);

    // S' = Phi*S + K*X         (6 chained WMMAs)
    v8f sn = {};
    sn = wmma4(PhiA[0], sb0, sn);
    sn = wmma4(PhiA[1], sb1, sn);
    for (int c = 0; c < 4; ++c) sn = wmma4(KmA[c], xb[c], sn);
    s0 = sn[0]; s1 = sn[1]; s2 = sn[2]; s3 = sn[3]; s4 = sn[4]; s5 = sn[5];

    // transpose through LDS for coalesced 128b stores along time
    for (int v = 0; v < 8; ++v) tile[n * 20 + v + 8 * h] = y[v];
    __syncthreads();
    for (int r = 0; r < 2; ++r) {
      const int idx = r * 32 + lane;
      const int row = idx >> 2;   // batch row in tile
      const int tq  = idx & 3;    // time quad
      const float4 val = *reinterpret_cast<const float4*>(&tile[row * 20 + tq * 4]);
      float* op = out + (size_t)(bt * 16 + row) * (NBANDS * TLEN) + (size_t)f * TLEN
                      + chunk * CHUNK_LEN + blk * BLK + tq * 4;
      *reinterpret_cast<float4*>(op) = val;
    }
    __syncthreads();
  }
}

// ---------------- host launch ----------------
extern "C" void kernel_launch(void* const* d_in, const int* in_sizes, int n_in,
                              void* d_out, int out_size, void* d_ws, size_t ws_size,
                              hipStream_t stream) {
  const float* x   = (const float*)d_in[0];   // [64, 96000]
  const float* sos = (const float*)d_in[1];   // [6, 3, 6]
  float* out = (float*)d_out;                 // [64, 6, 96000]
  float* ws  = (float*)d_ws;                  // needs ~1.9 MB

  sos_setup_kernel<<<1, 32, 0, stream>>>(sos, ws);
  phase1_kernel<<<dim3(CHUNKS, BTILES, NBANDS), 32, 0, stream>>>(x, ws);
  phase2_kernel<<<dim3(BTILES, NBANDS), 32, 0, stream>>>(ws);
  phase3_kernel<<<dim3(CHUNKS, BTILES, NBANDS), 32, 0, stream>>>(x, ws, out);
}